// Q0DistKf_41188736368929
// MI455X (gfx1250) — compile-verified
//
#include <hip/hip_runtime.h>
#include <math.h>

// ---------------- problem constants ----------------
constexpr int   N   = 336;
constexpr int   L   = 24;
constexpr int   C   = 16;
constexpr int   R   = 4;
constexpr int   S   = 100;
constexpr int   B   = 32;
constexpr int   NF  = N / 2 + 1;    // 169 rfft bins
constexpr int   CL  = C * L;        // 384
constexpr int   BC  = B * C;        // 512
constexpr float PI_F = 3.14159265358979323846f;
constexpr float W0   = PI_F / 24.0f;   // pi / FREQ  (GAMMA = 1)
constexpr float ISO  = 1e-4f;

// ---------------- workspace layout (float offsets) ----------------
constexpr size_t oKX   = 0;                       // 336  : kx_row
constexpr size_t oCOS  = 512;                     // 336  : cos(2*pi*m/N)
constexpr size_t oSIN  = 1024;                    // 336  : sin(2*pi*m/N)
constexpr size_t oREIG = 1536;                    // 169  : Re(rfft(kx_row))
constexpr size_t oKSX  = 2048;                    // 336*24 : K_star_x
constexpr size_t oKSS  = oKSX + (size_t)N * L;    // 576  : K_star_star
constexpr size_t oLAM  = oKSS + (size_t)L * L;    // 4    : top-R eigvals of Kf
constexpr size_t oU4   = oLAM + 8;                // 64   : Uf_r (16x4, row-major c*4+r)
constexpr size_t oUU   = oU4 + 64;                // 256  : Uf_r Uf_r^T
constexpr size_t oLOC  = oUU + 256;               // 512  : per-(b,c) mean of x
constexpr size_t oYFR  = oLOC + 512;              // 512*169 : Re Y_fft
constexpr size_t oYFI  = oYFR + (size_t)BC * NF;  // 512*169 : Im Y_fft
constexpr size_t oAHR  = oYFI + (size_t)BC * NF;  // 512*169 : Re Ah_fft
constexpr size_t oAHI  = oAHR + (size_t)BC * NF;  // 512*169 : Im Ah_fft
constexpr size_t oAW   = oAHI + (size_t)BC * NF;  // 512*336 : Aw (irfft)
constexpr size_t oMEAN = oAW  + (size_t)BC * N;   // 32*384  : mean
constexpr size_t oP    = oMEAN + (size_t)B * CL;  // 4*336*336 : lam_r*Kx + ISO (-> Chol L)
constexpr size_t oW    = oP   + (size_t)4 * N * N;// 4*336*24  : (lam Kx+ISO)^-1 Ksx
constexpr size_t oMP   = oW   + (size_t)4 * N * L;// 4*24*24   : M_r -> clamped M_r+
constexpr size_t oCOV  = oMP  + (size_t)4 * L * L;// 384*384   : cov -> Cholesky L

// ---------------- stage 0: constant tables ----------------
__global__ void k_const(float* ws) {
  int t = blockIdx.x * blockDim.x + threadIdx.x;
  if (t < N) {                                    // tables + kx_row
    float ang = 2.0f * PI_F * (float)t / (float)N;
    ws[oCOS + t] = cosf(ang);
    ws[oSIN + t] = sinf(ang);
    float d = (float)t * W0;
    ws[oKX + t] = expf(-d * d);
  }
  int t2 = t - N;
  if (t2 >= 0 && t2 < NF) {                       // Re rfft(kx_row), direct
    float acc = 0.0f;
    for (int n = 0; n < N; ++n) {
      float d  = (float)n * W0;
      int   m  = (t2 * n) % N;
      acc += expf(-d * d) * cosf(2.0f * PI_F * (float)m / (float)N);
    }
    ws[oREIG + t2] = acc;
  }
  int t3 = t - N - NF;
  if (t3 >= 0 && t3 < N * L) {                    // K_star_x[n][l]
    int n = t3 / L, l = t3 % L;
    float d = (float)(N + l - n) * W0;
    ws[oKSX + t3] = expf(-d * d);
  }
  int t4 = t - N - NF - N * L;
  if (t4 >= 0 && t4 < L * L) {                    // K_star_star
    int i = t4 / L, j = t4 % L;
    float d = (float)(i - j) * W0;
    ws[oKSS + t4] = expf(-d * d) + (i == j ? ISO : 0.0f);
  }
}

// ---------------- generic serial Jacobi eigensolver ----------------
template <int M>
__device__ void jacobi_sym(float (*A)[M], float (*V)[M], int sweeps) {
  for (int s = 0; s < sweeps; ++s) {
    for (int p = 0; p < M - 1; ++p) {
      for (int q = p + 1; q < M; ++q) {
        float apq = A[p][q];
        if (fabsf(apq) <= 1e-12f) continue;
        float theta = 0.5f * (A[q][q] - A[p][p]) / apq;
        float tt = 1.0f / (fabsf(theta) + sqrtf(1.0f + theta * theta));
        if (theta < 0.0f) tt = -tt;
        float cs = 1.0f / sqrtf(1.0f + tt * tt);
        float sn = tt * cs;
        for (int k = 0; k < M; ++k) {
          float akp = A[k][p], akq = A[k][q];
          A[k][p] = cs * akp - sn * akq;
          A[k][q] = sn * akp + cs * akq;
        }
        for (int k = 0; k < M; ++k) {
          float apk = A[p][k], aqk = A[q][k];
          A[p][k] = cs * apk - sn * aqk;
          A[q][k] = sn * apk + cs * aqk;
        }
        for (int k = 0; k < M; ++k) {
          float vkp = V[k][p], vkq = V[k][q];
          V[k][p] = cs * vkp - sn * vkq;
          V[k][q] = sn * vkp + cs * vkq;
        }
      }
    }
  }
}

// ---------------- stage 1: eigh(Kf + 1e-6 I), keep top R ----------------
__global__ void k_eigh16(const float* Kf, float* ws) {
  if (threadIdx.x != 0 || blockIdx.x != 0) return;
  float A[16][16], V[16][16];
  for (int i = 0; i < 16; ++i)
    for (int j = 0; j < 16; ++j) {
      A[i][j] = Kf[i * 16 + j] + (i == j ? 1e-6f : 0.0f);
      V[i][j] = (i == j) ? 1.0f : 0.0f;
    }
  jacobi_sym<16>(A, V, 30);
  float w[16];
  for (int i = 0; i < 16; ++i) w[i] = A[i][i];
  int  idx[4];
  bool used[16] = {false};
  for (int r = 0; r < 4; ++r) {
    int best = 0; float bv = -1e30f;
    for (int i = 0; i < 16; ++i)
      if (!used[i] && w[i] > bv) { bv = w[i]; best = i; }
    used[best] = true; idx[r] = best;
  }
  for (int r = 0; r < 4; ++r) {
    ws[oLAM + r] = w[idx[r]];
    for (int c = 0; c < 16; ++c) ws[oU4 + c * 4 + r] = V[c][idx[r]];
  }
  for (int c = 0; c < 16; ++c)
    for (int cc = 0; cc < 16; ++cc) {
      float s = 0.0f;
      for (int r = 0; r < 4; ++r) s += V[c][idx[r]] * V[cc][idx[r]];
      ws[oUU + c * 16 + cc] = s;
    }
}

// ---------------- stage 2: mean path (DFT based) ----------------
__global__ void k_loc(const float* x, float* ws) {
  int t = blockIdx.x * blockDim.x + threadIdx.x;
  if (t >= BC) return;
  const float* xr = x + (size_t)t * N;
  float s = 0.0f;
  for (int n = 0; n < N; ++n) s += xr[n];
  ws[oLOC + t] = s / (float)N;
}

__global__ void k_dft(const float* x, float* ws) {
  int t = blockIdx.x * blockDim.x + threadIdx.x;
  if (t >= BC * NF) return;
  int bc = t / NF, k = t % NF;
  const float* xr = x + (size_t)bc * N;
  const float* ct = ws + oCOS;
  const float* st = ws + oSIN;
  float loc = ws[oLOC + bc];
  float re = 0.0f, im = 0.0f;
  int m = 0;
  for (int n = 0; n < N; ++n) {
    float y = xr[n] - loc;
    re += y * ct[m];
    im -= y * st[m];
    m += k; if (m >= N) m -= N;
  }
  ws[oYFR + t] = re;
  ws[oYFI + t] = im;
}

__global__ void k_proj(float* ws) {
  int t = blockIdx.x * blockDim.x + threadIdx.x;
  if (t >= B * NF) return;
  int b = t / NF, k = t % NF;
  float reig = ws[oREIG + k];
  float ahr[16], ahi[16];
  for (int c = 0; c < 16; ++c) { ahr[c] = 0.0f; ahi[c] = 0.0f; }
  for (int r = 0; r < 4; ++r) {
    float ypr = 0.0f, ypi = 0.0f;
    for (int c = 0; c < 16; ++c) {
      float u = ws[oU4 + c * 4 + r];
      ypr += u * ws[oYFR + (size_t)(b * 16 + c) * NF + k];
      ypi += u * ws[oYFI + (size_t)(b * 16 + c) * NF + k];
    }
    float den = ws[oLAM + r] * reig + ISO;
    ypr /= den; ypi /= den;
    for (int c = 0; c < 16; ++c) {
      float u = ws[oU4 + c * 4 + r];
      ahr[c] += u * ypr; ahi[c] += u * ypi;
    }
  }
  for (int c = 0; c < 16; ++c) {
    ws[oAHR + (size_t)(b * 16 + c) * NF + k] = ahr[c];
    ws[oAHI + (size_t)(b * 16 + c) * NF + k] = ahi[c];
  }
}

__global__ void k_irfft(float* ws) {
  int t = blockIdx.x * blockDim.x + threadIdx.x;
  if (t >= BC * N) return;
  int bc = t / N, n = t % N;
  const float* rr = ws + oAHR + (size_t)bc * NF;
  const float* ii = ws + oAHI + (size_t)bc * NF;
  const float* ct = ws + oCOS;
  const float* st = ws + oSIN;
  float s = 0.0f;
  int m = 0;
  for (int k = 0; k < NF; ++k) {
    float wk = (k == 0 || k == NF - 1) ? 1.0f : 2.0f;
    s += wk * (rr[k] * ct[m] - ii[k] * st[m]);
    m += n; if (m >= N) m -= N;
  }
  ws[oAW + t] = s / (float)N;
}

__global__ void k_mean(float* ws) {
  int t = blockIdx.x * blockDim.x + threadIdx.x;
  if (t >= BC * L) return;
  int bc = t / L, l = t % L;
  int b = bc / 16, c = bc % 16;
  float s = 0.0f;
  for (int n = 0; n < N; ++n) s += ws[oAW + (size_t)bc * N + n] * ws[oKSX + n * L + l];
  ws[oMEAN + (size_t)b * CL + c * L + l] = s + ws[oLOC + bc];
}

// ---------------- stage 3: covariance path ----------------
__global__ void k_formP(float* ws) {
  int t = blockIdx.x * blockDim.x + threadIdx.x;
  if (t >= 4 * N * N) return;
  int r = t / (N * N), rem = t % (N * N);
  int i = rem / N, j = rem % N;
  int d = i > j ? i - j : j - i;
  ws[oP + t] = ws[oLAM + r] * ws[oKX + d] + (i == j ? ISO : 0.0f);
}

__device__ void chol_inplace(float* A, int n, float* col) {
  for (int k = 0; k < n; ++k) {
    __syncthreads();
    if (threadIdx.x == 0) A[(size_t)k * n + k] = sqrtf(A[(size_t)k * n + k]);
    __syncthreads();
    float dkk = A[(size_t)k * n + k];
    for (int i = k + 1 + threadIdx.x; i < n; i += blockDim.x) {
      float v = A[(size_t)i * n + k] / dkk;
      A[(size_t)i * n + k] = v;
      col[i] = v;
    }
    __syncthreads();
    for (int j = k + 1 + threadIdx.x; j < n; j += blockDim.x) {
      float ljk = col[j];
      for (int i = j; i < n; ++i) A[(size_t)i * n + j] -= col[i] * ljk;
    }
  }
  __syncthreads();
  for (int t = threadIdx.x; t < n * n; t += blockDim.x) {
    int i = t / n, j = t % n;
    if (j > i) A[t] = 0.0f;
  }
}

__global__ void k_chol336(float* ws) {
  __shared__ float col[N];
  chol_inplace(ws + oP + (size_t)blockIdx.x * N * N, N, col);
}

__global__ void k_solve(float* ws) {
  int r = blockIdx.x;
  int l = threadIdx.x;
  if (l >= L) return;
  const float* Lm = ws + oP + (size_t)r * N * N;
  float*       Wm = ws + oW + (size_t)r * N * L;
  for (int i = 0; i < N; ++i) {                   // forward: L y = Ksx[:,l]
    float s = ws[oKSX + i * L + l];
    for (int j = 0; j < i; ++j) s -= Lm[(size_t)i * N + j] * Wm[(size_t)j * L + l];
    Wm[(size_t)i * L + l] = s / Lm[(size_t)i * N + i];
  }
  for (int i = N - 1; i >= 0; --i) {              // backward: L^T w = y
    float s = Wm[(size_t)i * L + l];
    for (int j = i + 1; j < N; ++j) s -= Lm[(size_t)j * N + i] * Wm[(size_t)j * L + l];
    Wm[(size_t)i * L + l] = s / Lm[(size_t)i * N + i];
  }
}

__global__ void k_formM(float* ws) {
  int t = blockIdx.x * blockDim.x + threadIdx.x;
  if (t >= 4 * L * L) return;
  int r = t / (L * L), ij = t % (L * L);
  int i = ij / L, j = ij % L;
  float lam = ws[oLAM + r];
  float g = 0.0f;
  for (int n = 0; n < N; ++n)
    g += ws[oKSX + n * L + i] * ws[oW + (size_t)r * N * L + n * L + j];
  ws[oMP + t] = lam * ws[oKSS + ij] - lam * lam * g;
}

__global__ void k_eigh24(float* ws) {
  if (threadIdx.x != 0) return;
  int r = blockIdx.x;
  float* M0 = ws + oMP + (size_t)r * L * L;
  float A[24][24], V[24][24];
  for (int i = 0; i < 24; ++i)
    for (int j = 0; j < 24; ++j) {
      A[i][j] = 0.5f * (M0[i * L + j] + M0[j * L + i]);
      V[i][j] = (i == j) ? 1.0f : 0.0f;
    }
  jacobi_sym<24>(A, V, 40);
  float mu[24];
  for (int i = 0; i < 24; ++i) mu[i] = fmaxf(A[i][i], 1e-3f);
  for (int i = 0; i < 24; ++i)
    for (int j = 0; j < 24; ++j) {
      float s = 0.0f;
      for (int k = 0; k < 24; ++k) s += V[i][k] * mu[k] * V[j][k];
      M0[i * L + j] = s;
    }
}

__global__ void k_cov(float* ws) {
  int t = blockIdx.x * blockDim.x + threadIdx.x;
  if (t >= CL * CL) return;
  int row = t / CL, col = t % CL;
  int c = row / L, i = row % L;
  int cc = col / L, j = col % L;
  float s = 0.0f;
  for (int r = 0; r < 4; ++r)
    s += ws[oU4 + c * 4 + r] * ws[oU4 + cc * 4 + r] * ws[oMP + (size_t)r * L * L + i * L + j];
  if (i == j) s += 0.002f * (c == cc ? 1.0f : 0.0f) - 0.001f * ws[oUU + c * 16 + cc];
  ws[oCOV + t] = s;
}

__global__ void k_chol384(float* ws) {
  __shared__ float col[CL];
  chol_inplace(ws + oCOV, CL, col);
}

// ---------------- stage 4: samples = mean + eps @ L^T via WMMA f32 ----------------
typedef float v2f __attribute__((ext_vector_type(2)));
typedef float v8f __attribute__((ext_vector_type(8)));

__global__ void __launch_bounds__(32) k_sample(const float* eps, const float* ws, float* out) {
  int tn   = blockIdx.x;            // N-tile: 0..23
  int tm   = blockIdx.y;            // M-tile: 0..199
  int lane = threadIdx.x;
  int half = lane >> 4;             // K half-select per ISA A/B 32-bit layout
  int q    = lane & 15;             // A-row / B-col within tile
  const float* arow = eps            + (size_t)(tm * 16 + q) * CL + 2 * half;
  const float* brow = (ws + oCOV)    + (size_t)(tn * 16 + q) * CL + 2 * half; // Lchol row f
  v8f acc = {0.f, 0.f, 0.f, 0.f, 0.f, 0.f, 0.f, 0.f};
#pragma unroll 4
  for (int k0 = 0; k0 < CL; k0 += 4) {
    v2f a, b;
    a[0] = arow[k0]; a[1] = arow[k0 + 1];         // A[m][k0+2h], A[m][k0+2h+1]
    b[0] = brow[k0]; b[1] = brow[k0 + 1];         // B[k0+2h][n] = L[n][k0+2h]
    acc = __builtin_amdgcn_wmma_f32_16x16x4_f32(false, a, false, b,
                                                (short)0, acc, false, false);
  }
  int colg = tn * 16 + q;
#pragma unroll
  for (int v = 0; v < 8; ++v) {
    int m = tm * 16 + half * 8 + v;               // C/D layout: M = v + 8*(lane>>4)
    out[(size_t)m * CL + colg] = acc[v] + ws[oMEAN + (size_t)(m & 31) * CL + colg];
  }
}

// ---------------- host entry ----------------
extern "C" void kernel_launch(void* const* d_in, const int* in_sizes, int n_in,
                              void* d_out, int out_size, void* d_ws, size_t ws_size,
                              hipStream_t stream) {
  (void)in_sizes; (void)n_in; (void)out_size; (void)ws_size;
  const float* x   = (const float*)d_in[0];
  const float* Kf  = (const float*)d_in[1];
  const float* eps = (const float*)d_in[2];
  float* out = (float*)d_out;
  float* ws  = (float*)d_ws;

  k_const  <<<(N + NF + N * L + L * L + 255) / 256, 256, 0, stream>>>(ws);
  k_eigh16 <<<1, 32, 0, stream>>>(Kf, ws);
  k_loc    <<<(BC + 255) / 256, 256, 0, stream>>>(x, ws);
  k_dft    <<<(BC * NF + 255) / 256, 256, 0, stream>>>(x, ws);
  k_proj   <<<(B * NF + 255) / 256, 256, 0, stream>>>(ws);
  k_irfft  <<<(BC * N + 255) / 256, 256, 0, stream>>>(ws);
  k_mean   <<<(BC * L + 255) / 256, 256, 0, stream>>>(ws);
  k_formP  <<<(4 * N * N + 255) / 256, 256, 0, stream>>>(ws);
  k_chol336<<<4, 256, 0, stream>>>(ws);
  k_solve  <<<4, 32, 0, stream>>>(ws);
  k_formM  <<<(4 * L * L + 255) / 256, 256, 0, stream>>>(ws);
  k_eigh24 <<<4, 32, 0, stream>>>(ws);
  k_cov    <<<(CL * CL + 255) / 256, 256, 0, stream>>>(ws);
  k_chol384<<<1, 256, 0, stream>>>(ws);
  dim3 g(CL / 16, (S * B) / 16);                  // 24 x 200 tiles
  k_sample <<<g, 32, 0, stream>>>(eps, ws, out);
}